// SeesawLoss_prior_38671885533685
// MI455X (gfx1250) — compile-verified
//
#include <hip/hip_runtime.h>
#include <hip/hip_bf16.h>
#include <math.h>

// ---------------- problem constants ----------------
#define CLS            100
#define BATCH          262144
#define ROWS_PER_CHUNK 32
#define CHUNK_FLOATS   (ROWS_PER_CHUNK * CLS)      // 3200 floats = 12800 B (16B aligned)
#define NCHUNKS        (BATCH / ROWS_PER_CHUNK)    // 8192
#define BLOCKS         1024
#define THREADS        128
#define WAVES_PER_BLK  (THREADS / 32)              // 4
#define TOTAL_WAVES    (BLOCKS * WAVES_PER_BLK)    // 4096

#define AS1 __attribute__((address_space(1)))
#define AS3 __attribute__((address_space(3)))

typedef int vi4 __attribute__((vector_size(16))); // matches builtin param type

#ifndef __has_builtin
#define __has_builtin(x) 0
#endif

// gfx12 CPol: TH in bits[2:0]; TH=1 => non-temporal (x is streamed exactly once,
// keep the L2-resident log-weight table from being evicted)
#define CPOL_NT 1

// ---- CDNA5 async global->LDS DMA: 25 x b128 (32 lanes x 16B) = one 32-row chunk ----
__device__ __forceinline__ void stage_chunk_async(const float* __restrict__ gsrc,
                                                  float* ldst, int lane) {
#if __has_builtin(__builtin_amdgcn_global_load_async_to_lds_b128)
  AS1 vi4* g = (AS1 vi4*)gsrc + lane;   // +16B per element
  AS3 vi4* l = (AS3 vi4*)ldst + lane;
#pragma unroll
  for (int i = 0; i < 25; ++i) {
    __builtin_amdgcn_global_load_async_to_lds_b128(g + i * 32, l + i * 32, 0, CPOL_NT);
  }
#else
  // synchronous fallback (still correct, just not async)
#pragma unroll
  for (int i = 0; i < 25; ++i) {
    const float4 v = *reinterpret_cast<const float4*>(gsrc + i * 128 + lane * 4);
    *reinterpret_cast<float4*>(ldst + i * 128 + lane * 4) = v;
  }
#endif
}

template <int N>
__device__ __forceinline__ void wait_asynccnt_le() {
#if __has_builtin(__builtin_amdgcn_s_wait_asynccnt)
  __builtin_amdgcn_s_wait_asynccnt(N);
#else
  asm volatile("s_wait_asynccnt %0" ::"i"(N));
#endif
  asm volatile("" ::: "memory");   // keep LDS reads below the wait
}

// ---------------- LDS layout: 143,204 B / block ----------------
struct __align__(16) SeesawSmem {
  float logw[CLS * CLS];                          // 40,000 B  log(weight_matrix)
  float stage[WAVES_PER_BLK][2][CHUNK_FLOATS];    // 102,400 B per-wave double buffers
  float bneg[CLS];
  float bpos[CLS];
  float bloss;
};

__global__ __launch_bounds__(THREADS) void seesaw_main(
    const float* __restrict__ x, const int* __restrict__ target,
    const float* __restrict__ wm, float* __restrict__ acc) {
  __shared__ SeesawSmem sm;
  const int tid  = threadIdx.x;
  const int lane = tid & 31;
  const int wid  = tid >> 5;

  if (tid < CLS) { sm.bneg[tid] = 0.f; sm.bpos[tid] = 0.f; }
  if (tid == 0)  sm.bloss = 0.f;
  for (int i = tid; i < CLS * CLS; i += THREADS) sm.logw[i] = __logf(wm[i]);
  __syncthreads();

  float negacc[4] = {0.f, 0.f, 0.f, 0.f};
  float posacc[4] = {0.f, 0.f, 0.f, 0.f};
  float lossacc   = 0.f;

  float* buf0 = sm.stage[wid][0];
  float* buf1 = sm.stage[wid][1];

  const int wgl = blockIdx.x * WAVES_PER_BLK + wid;   // global wave id, 0..4095
  int cur = 0;
  if (wgl < NCHUNKS) stage_chunk_async(x + (size_t)wgl * CHUNK_FLOATS, buf0, lane);

  for (int chunk = wgl; chunk < NCHUNKS; chunk += TOTAL_WAVES) {
    const int nxt = chunk + TOTAL_WAVES;
    float* curBuf = cur ? buf1 : buf0;
    float* nxtBuf = cur ? buf0 : buf1;
    if (nxt < NCHUNKS) {
      stage_chunk_async(x + (size_t)nxt * CHUNK_FLOATS, nxtBuf, lane);
      wait_asynccnt_le<25>();     // in-order: first 25 (current chunk) have landed
    } else {
      wait_asynccnt_le<0>();
    }
    const int tv = target[chunk * ROWS_PER_CHUNK + lane];  // 32 targets, one per lane

#pragma unroll 2                  // interleave two rows' reduction chains (ILP)
    for (int j = 0; j < ROWS_PER_CHUNK; ++j) {
      const int t = __shfl(tv, j, 32);
      const float* lw = sm.logw + t * CLS;
      const float* xr = curBuf + j * CLS;

      float v[4], e[4];
      float m = -__builtin_inff();
#pragma unroll
      for (int k = 0; k < 4; ++k) {
        const int c = lane + 32 * k;
        v[k] = (c < CLS) ? (xr[c] + lw[c]) : -__builtin_inff();
        m = fmaxf(m, v[k]);
      }
#pragma unroll
      for (int off = 16; off >= 1; off >>= 1)
        m = fmaxf(m, __shfl_xor(m, off, 32));

      float s = 0.f;
#pragma unroll
      for (int k = 0; k < 4; ++k) { e[k] = __expf(v[k] - m); s += e[k]; }
#pragma unroll
      for (int off = 16; off >= 1; off >>= 1)
        s += __shfl_xor(s, off, 32);

      const float inv = __builtin_amdgcn_rcpf(s);
      const float ls  = __logf(s);
      const int tl = t & 31, tk = t >> 5;
#pragma unroll
      for (int k = 0; k < 4; ++k) {
        const float p = e[k] * inv;       // softmax_c (0 for padded cols)
        negacc[k] += p;
        if (lane == tl && k == tk) {      // c == target
          negacc[k] -= p;                 // neg excludes target class
          posacc[k] += 1.f - p;           // |softmax_t - 1|
          lossacc   += (m + ls) - v[k];   // -log_softmax at target
        }
      }
    }
    cur ^= 1;
  }

  // block-level reduction in LDS, then one global push per block
#pragma unroll
  for (int k = 0; k < 4; ++k) {
    const int c = lane + 32 * k;
    if (c < CLS) {
      atomicAdd(&sm.bneg[c], negacc[k]);
      atomicAdd(&sm.bpos[c], posacc[k]);
    }
  }
#pragma unroll
  for (int off = 16; off >= 1; off >>= 1)
    lossacc += __shfl_xor(lossacc, off, 32);
  if (lane == 0) atomicAdd(&sm.bloss, lossacc);
  __syncthreads();

  if (tid < CLS) {
    atomicAdd(&acc[tid], sm.bneg[tid]);
    atomicAdd(&acc[CLS + tid], sm.bpos[tid]);
  }
  if (tid == 0) atomicAdd(&acc[2 * CLS], sm.bloss);
}

__global__ void seesaw_init(float* __restrict__ acc) {
  const int i = threadIdx.x;
  if (i < 2 * CLS + 1) acc[i] = 0.f;
}

__global__ void seesaw_finalize(const float* __restrict__ acc,
                                const float* __restrict__ neg_in,
                                const float* __restrict__ pos_in,
                                float* __restrict__ out) {
  const int c = threadIdx.x;
  if (c == 0) out[0] = acc[2 * CLS] * (1.0f / (float)BATCH);
  if (c < CLS) {
    const float nn = neg_in[c] + acc[c];
    const float pp = pos_in[c] + acc[CLS + c];
    out[1 + c]           = nn;
    out[1 + CLS + c]     = pp;
    out[1 + 2 * CLS + c] = pp / nn;
  }
}

extern "C" void kernel_launch(void* const* d_in, const int* in_sizes, int n_in,
                              void* d_out, int out_size, void* d_ws, size_t ws_size,
                              hipStream_t stream) {
  (void)in_sizes; (void)n_in; (void)out_size; (void)ws_size;
  const float* x      = (const float*)d_in[0];
  const int*   target = (const int*)d_in[1];
  const float* wm     = (const float*)d_in[2];
  const float* ng     = (const float*)d_in[3];
  const float* pg     = (const float*)d_in[4];
  float* out = (float*)d_out;
  float* acc = (float*)d_ws;   // [neg 100][pos 100][loss 1]

  hipLaunchKernelGGL(seesaw_init,     dim3(1),      dim3(256),     0, stream, acc);
  hipLaunchKernelGGL(seesaw_main,     dim3(BLOCKS), dim3(THREADS), 0, stream, x, target, wm, acc);
  hipLaunchKernelGGL(seesaw_finalize, dim3(1),      dim3(128),     0, stream, acc, ng, pg, out);
}